// Encoder_6227702579879
// MI455X (gfx1250) — compile-verified
//
#include <hip/hip_runtime.h>

typedef __attribute__((ext_vector_type(2))) float v2f;
typedef __attribute__((ext_vector_type(8))) float v8f;
typedef __attribute__((ext_vector_type(4))) int   v4i;

#define DIMH 512
#define BATCH 32
#define TLEN 512
#define NTOK 14
#define G4 (4*DIMH)                        // 2048 gate rows

// ---- workspace layout (float offsets) ----
// P table: [2][14][2048]
#define OFF_P   0
// h double buffers: [2 lstm][2 buf][32][512]
#define OFF_H   (2*NTOK*G4)                        // 57344
// global barrier counter (one uint, padded)
#define OFF_CTR (OFF_H + 2*2*BATCH*DIMH)           // 122880
// hidden-state sequences hs_q, hs_k: [2][32][512][512]
#define OFF_HS  (OFF_CTR + 64)                     // 122944
#define HS_SZ   (BATCH*TLEN*DIMH)                  // 8388608 floats each

// gfx1250 async LDS-DMA path (ASYNCcnt-tracked), guarded for toolchain drift
#if defined(__has_builtin)
#  if __has_builtin(__builtin_amdgcn_global_load_async_to_lds_b128) && \
      __has_builtin(__builtin_amdgcn_s_wait_asynccnt)
#    define USE_ASYNC_LDS 1
#  endif
#endif

__device__ __forceinline__ float sigm(float x) { return 1.0f / (1.0f + __expf(-x)); }

// ---------------- init: zero h0 and barrier counter ----------------
__global__ void k_init(float* ws) {
    int i = blockIdx.x * blockDim.x + threadIdx.x;
    if (i < 2*2*BATCH*DIMH) ws[OFF_H + i] = 0.0f;
    if (i == 0) *(unsigned*)(ws + OFF_CTR) = 0u;
}

// ---------------- gate table: P[l][tok][g] = embed[tok]@W_ih^T + b_ih + b_hh ----
__global__ void k_gate_table(const float* __restrict__ embed,
                             const float* __restrict__ Wq, const float* __restrict__ bq1,
                             const float* __restrict__ bq2,
                             const float* __restrict__ Wk, const float* __restrict__ bk1,
                             const float* __restrict__ bk2,
                             float* __restrict__ ws) {
    int i = blockIdx.x * blockDim.x + threadIdx.x;     // 0 .. 2*14*2048-1
    int l   = i / (NTOK * G4);
    int r   = i % (NTOK * G4);
    int tok = r / G4;
    int g   = r % G4;
    const float* W  = l ? Wk : Wq;
    float bias      = l ? (bk1[g] + bk2[g]) : (bq1[g] + bq2[g]);
    const float4* e4 = (const float4*)(embed + tok * DIMH);
    const float4* w4 = (const float4*)(W + (size_t)g * DIMH);
    float s = 0.0f;
    for (int k = 0; k < DIMH/4; ++k) {
        float4 a = e4[k], b = w4[k];
        s += a.x*b.x + a.y*b.y + a.z*b.z + a.w*b.w;
    }
    ws[OFF_P + i] = s + bias;
}

// ---------------- device-wide release/acquire barrier (monotone ticket) --------
__device__ __forceinline__ void gbar(unsigned* ctr, unsigned target) {
    __threadfence();
    __syncthreads();
    if (threadIdx.x == 0) {
        __hip_atomic_fetch_add(ctr, 1u, __ATOMIC_ACQ_REL, __HIP_MEMORY_SCOPE_AGENT);
        while (__hip_atomic_load(ctr, __ATOMIC_ACQUIRE, __HIP_MEMORY_SCOPE_AGENT) < target) {
            __builtin_amdgcn_s_sleep(8);
        }
    }
    __syncthreads();
    __threadfence();
}

// ---------------- LSTM scan: 64 WGs (32 per LSTM), 16 hidden units per WG ------
// LDS: W_hh slice, gate-interleaved [unit 0..15][gate 0..3][512] f32 (128KB)
//      so each unit's i/f/g/o rows sit in one 8KB window -> DS immediate offsets.
// Plus P slice [14][64] (3.5KB), same interleave.
__global__ void __launch_bounds__(64, 1)
k_lstm_scan(const int* __restrict__ x,
            const float* __restrict__ Whh_q, const float* __restrict__ Whh_k,
            float* __restrict__ ws) {
    extern __shared__ float lds[];
    float* Wl = lds;                 // [64][512]: row = unit_local*4 + gate
    float* Pl = lds + 64*DIMH;       // [14][64]:  col = unit_local*4 + gate

    const int wg    = blockIdx.x;    // 0..63
    const int l     = wg >> 5;       // 0=q, 1=k
    const int slice = wg & 31;
    const int j0    = slice * 16;    // first hidden unit owned by this WG
    const int wave  = threadIdx.x >> 5;   // 0..1 : M-tile (batch 0-15 / 16-31)
    const int lane  = threadIdx.x & 31;

    const float* Whh = l ? Whh_k : Whh_q;

    // stage W_hh slice: one LDS row per thread (64 rows x 512 f32 = 128KB)
    {
        int r    = threadIdx.x;              // LDS row = ul*4 + g
        int ul   = r >> 2, g = r & 3;
        int grow = g * DIMH + j0 + ul;       // global W_hh row: gate*512 + unit
        const float* src = Whh + (size_t)grow * DIMH;
        float*       dst = Wl + (size_t)r * DIMH;
#ifdef USE_ASYNC_LDS
        typedef __attribute__((address_space(1))) v4i gv4i;
        typedef __attribute__((address_space(3))) v4i lv4i;
        gv4i* gs = (gv4i*)src;
        lv4i* ld = (lv4i*)dst;
        for (int k = 0; k < DIMH/4; ++k)
            __builtin_amdgcn_global_load_async_to_lds_b128(gs + k, ld + k, 0, 0);
        __builtin_amdgcn_s_wait_asynccnt(0);
#else
        const float4* s4 = (const float4*)src;
        float4*       d4 = (float4*)dst;
        for (int k = 0; k < DIMH/4; ++k) d4[k] = s4[k];
#endif
    }
    // stage P slice (gate-interleaved)
    for (int idx = threadIdx.x; idx < NTOK*64; idx += 64) {
        int tok = idx >> 6;
        int rr  = idx & 63;
        int ul  = rr >> 2, g = rr & 3;
        Pl[idx] = ws[OFF_P + (size_t)(l*NTOK + tok)*G4 + g*DIMH + j0 + ul];
    }
    __syncthreads();

    unsigned* ctr   = (unsigned*)(ws + OFF_CTR);
    float*    hbase = ws + OFF_H + (size_t)l*2*BATCH*DIMH;
    float*    hs    = ws + OFF_HS + (size_t)l*HS_SZ;

    const int m0   = wave * 16;
    const int arow = m0 + (lane & 15);       // batch row for A fragment
    const int koff = (lane >> 4) << 1;       // K sub-offset per WMMA f32 layout
    const int n    = lane & 15;              // unit column within tile
    const int jg   = j0 + n;                 // global hidden unit

    float c[8];
    #pragma unroll
    for (int r = 0; r < 8; ++r) c[r] = 0.0f;

    int cur = 0;
    for (int t = 0; t < TLEN; ++t) {
        const float* hcur = hbase + (size_t)cur*BATCH*DIMH;
        float*       hnxt = hbase + (size_t)(cur^1)*BATCH*DIMH;

        v8f acc0 = {}, acc1 = {}, acc2 = {}, acc3 = {};
        const float* ha = hcur + (size_t)arow*DIMH + koff;
        const float* wp = Wl + (size_t)(n*4)*DIMH + koff;   // unit n's 4 gate rows
        for (int kk = 0; kk < DIMH/4; ++kk) {
            v2f a  = *(const v2f*)(ha + kk*4);
            v2f b0 = *(const v2f*)(wp +           kk*4);    // i
            v2f b1 = *(const v2f*)(wp +   DIMH  + kk*4);    // f
            v2f b2 = *(const v2f*)(wp + 2*DIMH  + kk*4);    // g
            v2f b3 = *(const v2f*)(wp + 3*DIMH  + kk*4);    // o
            acc0 = __builtin_amdgcn_wmma_f32_16x16x4_f32(false, a, false, b0, (short)0, acc0, false, false);
            acc1 = __builtin_amdgcn_wmma_f32_16x16x4_f32(false, a, false, b1, (short)0, acc1, false, false);
            acc2 = __builtin_amdgcn_wmma_f32_16x16x4_f32(false, a, false, b2, (short)0, acc2, false, false);
            acc3 = __builtin_amdgcn_wmma_f32_16x16x4_f32(false, a, false, b3, (short)0, acc3, false, false);
        }
        // gate fusion: i/f/g/o tiles share (lane,VGPR) <-> (batch,unit) mapping
        #pragma unroll
        for (int r = 0; r < 8; ++r) {
            int M   = m0 + r + ((lane >> 4) << 3);     // batch
            int tok = x[M*TLEN + t];
            float iv = sigm (acc0[r] + Pl[tok*64 + n*4 + 0]);
            float fv = sigm (acc1[r] + Pl[tok*64 + n*4 + 1]);
            float gv = tanhf(acc2[r] + Pl[tok*64 + n*4 + 2]);
            float ov = sigm (acc3[r] + Pl[tok*64 + n*4 + 3]);
            c[r] = fv * c[r] + iv * gv;
            float hv = ov * tanhf(c[r]);
            hnxt[M*DIMH + jg]                  = hv;
            hs[((size_t)M*TLEN + t)*DIMH + jg] = hv;
        }
        gbar(ctr, 64u * (unsigned)(t + 1));
        cur ^= 1;
    }
}

// ---------------- A = q @ k^T per batch (fp32 WMMA) ----------------------------
__global__ void __launch_bounds__(256)
k_bmm(float* __restrict__ out, const float* __restrict__ ws) {
    int blk   = blockIdx.x;          // 0 .. 32*128-1
    int b     = blk >> 7;
    int wgid  = blk & 127;
    int mtile = wgid & 31;
    int ngrp  = wgid >> 5;           // 0..3
    int wave  = threadIdx.x >> 5;
    int lane  = threadIdx.x & 31;
    int ntile = ngrp * 8 + wave;

    const float* qb = ws + OFF_HS +         (size_t)(b*TLEN)*DIMH;
    const float* kb = ws + OFF_HS + HS_SZ + (size_t)(b*TLEN)*DIMH;
    int arow = mtile*16 + (lane & 15);
    int brow = ntile*16 + (lane & 15);
    int koff = (lane >> 4) << 1;
    const float* ap = qb + (size_t)arow*DIMH + koff;
    const float* bp = kb + (size_t)brow*DIMH + koff;

    v8f acc = {};
    for (int kk = 0; kk < DIMH/4; ++kk) {
        v2f a  = *(const v2f*)(ap + kk*4);
        v2f bb = *(const v2f*)(bp + kk*4);
        acc = __builtin_amdgcn_wmma_f32_16x16x4_f32(false, a, false, bb, (short)0, acc, false, false);
    }
    float* ob = out + (size_t)b*TLEN*TLEN;
    #pragma unroll
    for (int r = 0; r < 8; ++r) {
        int M = mtile*16 + r + ((lane >> 4) << 3);
        int N = ntile*16 + (lane & 15);
        ob[(size_t)M*TLEN + N] = acc[r];
    }
}

// ---------------- launcher ----------------------------------------------------
extern "C" void kernel_launch(void* const* d_in, const int* in_sizes, int n_in,
                              void* d_out, int out_size, void* d_ws, size_t ws_size,
                              hipStream_t stream) {
    const int*   x     = (const int*)d_in[0];
    const float* embed = (const float*)d_in[1];
    const float* Wih_q = (const float*)d_in[2];
    const float* Whh_q = (const float*)d_in[3];
    const float* bih_q = (const float*)d_in[4];
    const float* bhh_q = (const float*)d_in[5];
    const float* Wih_k = (const float*)d_in[6];
    const float* Whh_k = (const float*)d_in[7];
    const float* bih_k = (const float*)d_in[8];
    const float* bhh_k = (const float*)d_in[9];
    float* ws  = (float*)d_ws;
    float* out = (float*)d_out;

    k_init<<<(2*2*BATCH*DIMH + 255)/256, 256, 0, stream>>>(ws);
    k_gate_table<<<(2*NTOK*G4)/256, 256, 0, stream>>>(embed, Wih_q, bih_q, bhh_q,
                                                      Wih_k, bih_k, bhh_k, ws);
    size_t ldsz = (size_t)(64*DIMH + NTOK*64) * sizeof(float);   // ~134.7 KB
    k_lstm_scan<<<64, 64, ldsz, stream>>>(x, Whh_q, Whh_k, ws);
    k_bmm<<<32*128, 256, 0, stream>>>(out, ws);
}